// Conv2d_62826781606523
// MI455X (gfx1250) — compile-verified
//
#include <hip/hip_runtime.h>

typedef __attribute__((ext_vector_type(2))) float v2f;
typedef __attribute__((ext_vector_type(8))) float v8f;

namespace {
constexpr int kCIN = 16, kH = 32, kW = 32;
constexpr int kCOUT = 32, kHOUT = 30, kWOUT = 30;
constexpr int kP    = kHOUT * kWOUT;     // 900
constexpr int kF    = kCIN * 9;          // 144
constexpr int kNPix = kCOUT * kP;        // 28800
constexpr int kXB   = kCIN * kH * kW;    // 16384 floats per batch image
constexpr int kWavesPerBlock = 8;        // 256 threads = 8 wave32
}

__global__ __launch_bounds__(256) void lutconv_wmma_kernel(
    const float* __restrict__ x,        // [4,16,32,32]
    const float* __restrict__ weight,   // [T,4]
    const int*   __restrict__ mask,     // [2T,3]
    float*       __restrict__ out)      // [4,32,30,30]
{
  const int lane = threadIdx.x & 31;
  const int wave = threadIdx.x >> 5;
  const int pix  = blockIdx.x * kWavesPerBlock + wave;   // grid divides exactly

  const int half = lane >> 4;     // 0 -> batches {0,1}; 1 -> batches {2,3}
  const int col  = lane & 15;     // which of the 16 t's in this WMMA group

  // A (16x4) = constant +/-1 truth-table transform, rows replicated mod 4:
  //   row0: + + + +   row1: - - + +   row2: - + - +   row3: + - - +
  // A-matrix VGPR layout: lane supplies A[col][k] with k = 2*half + v.
  const int m4 = col & 3;
  v2f A;
  {
    const int k0 = 2 * half, k1 = k0 + 1;
    const float t0a = (k0 >= 2) ? 1.f : -1.f, t1a = (k0 & 1) ? 1.f : -1.f;
    const float t0b = (k1 >= 2) ? 1.f : -1.f, t1b = (k1 & 1) ? 1.f : -1.f;
    A[0] = (m4 == 0) ? 1.f : (m4 == 1) ? t0a : (m4 == 2) ? t1a : t0a * t1a;
    A[1] = (m4 == 0) ? 1.f : (m4 == 1) ? t0b : (m4 == 2) ? t1b : t0b * t1b;
  }

  const float* xb0 = x + (half * 2 + 0) * kXB;
  const float* xb1 = x + (half * 2 + 1) * kXB;
  const int2*  mask2 = (const int2*)mask;   // 2 rows * 3 ints = 3 x int2 per t

  float acc0 = 0.f, acc1 = 0.f;

  #pragma unroll 3
  for (int fb = 0; fb < kF; fb += 16) {
    const int t = pix * kF + fb + col;

    // B (4x16): lane supplies B[2*half+v][col] = weight[t*4 + 2*half + v].
    // Across the wave this is one contiguous 256B segment (16 t's x 16B).
    v2f Bv = *(const v2f*)(weight + t * 4 + 2 * half);

    v8f D = {0.f, 0.f, 0.f, 0.f, 0.f, 0.f, 0.f, 0.f};
    D = __builtin_amdgcn_wmma_f32_16x16x4_f32(
        /*neg_a=*/false, A, /*neg_b=*/false, Bv,
        /*c_mod=*/(short)0, D, /*reuse_a=*/false, /*reuse_b=*/false);
    // D VGPR r: lanes 0-15 -> row r (c0..c3), lanes 16-31 -> row r+8 which
    // replicates rows 0-3; column N == col for both halves.
    const float c0 = D[0], c1 = D[1], c2 = D[2], c3 = D[3];

    // Gather indices: mask rows 2t and 2t+1, each (c,h,w).
    const int2 ma = mask2[t * 3 + 0];
    const int2 mb = mask2[t * 3 + 1];
    const int2 mc = mask2[t * 3 + 2];
    const int i0 = ma.x * (kH * kW) + ma.y * kW + mb.x;
    const int i1 = mb.y * (kH * kW) + mc.x * kW + mc.y;

    const float e00 = xb0[i0], e01 = xb0[i1];
    const float e10 = xb1[i0], e11 = xb1[i1];
    acc0 += c0 + c1 * e00 + c2 * e01 + c3 * (e00 * e01);
    acc1 += c0 + c1 * e10 + c2 * e11 + c3 * (e10 * e11);
  }

  // Butterfly sum across the 16 lanes of each half (xor masks < 16 keep the
  // two batch-pair halves independent).
  #pragma unroll
  for (int m = 8; m >= 1; m >>= 1) {
    acc0 += __shfl_xor(acc0, m, 32);
    acc1 += __shfl_xor(acc1, m, 32);
  }

  if (col == 0) {
    out[(half * 2 + 0) * kNPix + pix] = acc0;
    out[(half * 2 + 1) * kNPix + pix] = acc1;
  }
}

extern "C" void kernel_launch(void* const* d_in, const int* in_sizes, int n_in,
                              void* d_out, int out_size, void* d_ws, size_t ws_size,
                              hipStream_t stream) {
  (void)in_sizes; (void)n_in; (void)out_size; (void)d_ws; (void)ws_size;
  const float* x      = (const float*)d_in[0];
  const float* weight = (const float*)d_in[1];
  const int*   mask   = (const int*)d_in[2];
  float*       out    = (float*)d_out;

  dim3 grid(kNPix / kWavesPerBlock);   // 3600 blocks
  dim3 block(kWavesPerBlock * 32);     // 256 threads = 8 waves
  lutconv_wmma_kernel<<<grid, block, 0, stream>>>(x, weight, mask, out);
}